// UnionSNNLayer_77214922047676
// MI455X (gfx1250) — compile-verified
//
#include <hip/hip_runtime.h>
#include <hip/hip_bf16.h>
#include <math.h>

// ---------------------------------------------------------------------------
// UnionSNN layer for MI455X (gfx1250, wave32).
//
// Key algorithmic collapse: the edge MLP input is rank-1 (weight[e] * w1_W),
// so leaky_relu is piecewise-linear in the scalar weight[e]:
//   ow[e,j] = weight[e] * (w_e>=0 ? a_pos[j] : a_neg[j]) + w2_b[j]
// with a_pos = w2_W @ lrelu(+w1), a_neg = w2_W @ lrelu(-side w1).
// Softmax over the edge axis cancels w2_b; the column max is attained at a
// weight extreme per sign group (4 scalars). This removes the E x 64 GEMM and
// all E x 64 intermediates; the layer becomes L2-resident gather/scatter
// bound (~410 MB against 12.8 MB tables, all inside the 192 MB L2).
// Node GEMMs use V_WMMA_F32_16X16X4_F32 (full fp32, matches reference).
// ---------------------------------------------------------------------------

typedef __attribute__((ext_vector_type(2))) float v2f;
typedef __attribute__((ext_vector_type(8))) float v8f;

#define N_NODES 50000
#define N_EDGES 800000
#define DIN 64
#define DOUT 64
#define HID 128
#define BN_EPS 1e-5f

// ws layout (floats):
//   [0, N*64)          neigh (scatter target, zeroed per call)
//   [N*64, 2N*64)      h1  (later overwritten in place by h3)
//   small block at 2N*64:
#define OFF_H1    ((size_t)N_NODES * DOUT)
#define OFF_SMALL ((size_t)2 * N_NODES * DOUT)
//   +0   a_pos[64], +64 a_neg[64], +128 colmax[64], +192 colsum[64]
//   +256 gsum[64],  +320 gsumsq[64], +384 enc[4] (uint: maxNN,minNN,maxNeg,minNeg)

static __device__ __forceinline__ unsigned encf(float f) {
  unsigned b = __float_as_uint(f);
  return (b & 0x80000000u) ? ~b : (b | 0x80000000u);  // order-preserving
}
static __device__ __forceinline__ float decf(unsigned u) {
  return __uint_as_float((u & 0x80000000u) ? (u ^ 0x80000000u) : ~u);
}
static __device__ __forceinline__ unsigned umax32(unsigned a, unsigned b) { return a > b ? a : b; }
static __device__ __forceinline__ unsigned umin32(unsigned a, unsigned b) { return a < b ? a : b; }

// ---------------- init: zero neigh + accumulators, seed extreme sentinels ---
__global__ void k_init(float* ws) {
  size_t stride = (size_t)gridDim.x * blockDim.x;
  for (size_t p = (size_t)blockIdx.x * blockDim.x + threadIdx.x;
       p < (size_t)N_NODES * DOUT; p += stride)
    ws[p] = 0.0f;  // neigh
  if (blockIdx.x == 0) {
    float* small = ws + OFF_SMALL;
    if (threadIdx.x < 192) small[192 + threadIdx.x] = 0.0f;  // colsum,gsum,gsumsq
    if (threadIdx.x == 0) {
      unsigned* enc = (unsigned*)(small + 384);
      enc[0] = 0u; enc[2] = 0u;                    // max accumulators
      enc[1] = 0xFFFFFFFFu; enc[3] = 0xFFFFFFFFu;  // min accumulators
    }
  }
}

// ---------------- weight extremes per sign group (for exact column max) ----
__global__ void __launch_bounds__(256) k_wred(const float* __restrict__ weight, float* ws) {
  unsigned mxnn = 0u, mnnn = 0xFFFFFFFFu, mxng = 0u, mnng = 0xFFFFFFFFu;
  for (int e = blockIdx.x * blockDim.x + threadIdx.x; e < N_EDGES;
       e += gridDim.x * blockDim.x) {
    float w = weight[e];
    unsigned u = encf(w);
    if (w >= 0.0f) { mxnn = umax32(mxnn, u); mnnn = umin32(mnnn, u); }
    else           { mxng = umax32(mxng, u); mnng = umin32(mnng, u); }
  }
#pragma unroll
  for (int off = 16; off >= 1; off >>= 1) {
    mxnn = umax32(mxnn, (unsigned)__shfl_xor((int)mxnn, off, 32));
    mnnn = umin32(mnnn, (unsigned)__shfl_xor((int)mnnn, off, 32));
    mxng = umax32(mxng, (unsigned)__shfl_xor((int)mxng, off, 32));
    mnng = umin32(mnng, (unsigned)__shfl_xor((int)mnng, off, 32));
  }
  if ((threadIdx.x & 31) == 0) {
    unsigned* enc = (unsigned*)(ws + OFF_SMALL + 384);
    atomicMax(enc + 0, mxnn); atomicMin(enc + 1, mnnn);
    atomicMax(enc + 2, mxng); atomicMin(enc + 3, mnng);
  }
}

// ---------------- a_pos/a_neg (64-vec each) + analytic column max ----------
__global__ void k_prep(const float* __restrict__ w1_W, const float* __restrict__ w2_W,
                       float* ws) {
  float* small = ws + OFF_SMALL;
  int j = threadIdx.x;  // 0..63
  float ap = 0.0f, an = 0.0f;
  for (int d = 0; d < DOUT; ++d) {
    float w1 = w1_W[d];
    float lp = (w1 >= 0.0f) ? w1 : 0.2f * w1;   // lrelu for w_e >= 0 branch
    float ln = (w1 <  0.0f) ? w1 : 0.2f * w1;   // lrelu for w_e <  0 branch
    float w2 = w2_W[j * DOUT + d];
    ap += w2 * lp; an += w2 * ln;
  }
  small[j] = ap; small[64 + j] = an;
  const unsigned* enc = (const unsigned*)(small + 384);
  float cm = -INFINITY;
  if (enc[0] != 0u) {  // nonneg-weight group exists; linear in w_e -> endpoints
    float mx = decf(enc[0]), mn = decf(enc[1]);
    cm = fmaxf(cm, fmaxf(mx * ap, mn * ap));
  }
  if (enc[2] != 0u) {
    float mx = decf(enc[2]), mn = decf(enc[3]);
    cm = fmaxf(cm, fmaxf(mx * an, mn * an));
  }
  small[128 + j] = cm;
}

// ---------------- h1 = h @ lin_W^T + lin_b via V_WMMA_F32_16X16X4_F32 ------
__global__ void __launch_bounds__(128) k_lin(const float* __restrict__ h,
                                             const float* __restrict__ lin_W,
                                             const float* __restrict__ lin_b,
                                             float* ws) {
  float* h1 = ws + OFF_H1;
  int wid  = blockIdx.x * 4 + (threadIdx.x >> 5);
  int lane = threadIdx.x & 31;
  int tileM = wid >> 2, tileN = wid & 3;     // 3125 x 4 tiles, exact fit
  int m   = lane & 15;
  int khi = (lane >= 16) ? 2 : 0;            // ISA f32 A/B fragment K split
  const float* arow = h     + (size_t)(tileM * 16 + m) * DIN;
  const float* brow = lin_W + (size_t)(tileN * 16 + m) * DIN;  // B[k][j]=lin_W[j][k]
  v8f c = {};
#pragma unroll
  for (int k0 = 0; k0 < DIN; k0 += 4) {
    v2f a = *(const v2f*)(arow + k0 + khi);
    v2f b = *(const v2f*)(brow + k0 + khi);
    c = __builtin_amdgcn_wmma_f32_16x16x4_f32(false, a, false, b, (short)0, c,
                                              false, false);
  }
  int col   = tileN * 16 + m;
  int rbase = tileM * 16 + ((lane >= 16) ? 8 : 0);
  float bias = lin_b[col];
#pragma unroll
  for (int r = 0; r < 8; ++r)
    h1[(size_t)(rbase + r) * DOUT + col] = c[r] + bias;
}

// ---------------- softmax denominators: colsum[j] = sum_e exp(...) ---------
__global__ void __launch_bounds__(256) k_colsum(const float* __restrict__ weight,
                                                float* ws) {
  float* small = ws + OFF_SMALL;
  __shared__ float s_ap[64], s_an[64], s_cm[64];
  __shared__ float red[4][64];
  int tid = threadIdx.x;
  if (tid < 64) { s_ap[tid] = small[tid]; s_an[tid] = small[64 + tid]; s_cm[tid] = small[128 + tid]; }
  __syncthreads();
  int j = tid & 63, sub = tid >> 6;
  float acc = 0.0f;
  for (int e = blockIdx.x * 4 + sub; e < N_EDGES; e += gridDim.x * 4) {
    float we = weight[e];
    float a = (we >= 0.0f) ? s_ap[j] : s_an[j];
    acc += __expf(we * a - s_cm[j]);
  }
  red[sub][j] = acc;
  __syncthreads();
  if (tid < 64)
    atomicAdd(small + 192 + tid, red[0][tid] + red[1][tid] + red[2][tid] + red[3][tid]);
}

// ---------------- m = h1[src]*w ; neigh[dst] += m (L2-resident atomics) ----
__global__ void __launch_bounds__(256) k_scatter(const float* __restrict__ weight,
                                                 const int* __restrict__ src,
                                                 const int* __restrict__ dst,
                                                 float* ws) {
  float* neigh = ws;
  const float* h1 = ws + OFF_H1;
  float* small = ws + OFF_SMALL;
  __shared__ float s_ap[64], s_an[64], s_cm[64], s_inv[64];
  int tid = threadIdx.x;
  if (tid < 64) {
    s_ap[tid] = small[tid]; s_an[tid] = small[64 + tid]; s_cm[tid] = small[128 + tid];
    s_inv[tid] = 1.0f / small[192 + tid];
  }
  __syncthreads();
  int j = tid & 63, sub = tid >> 6;
  for (int e = blockIdx.x * 4 + sub; e < N_EDGES; e += gridDim.x * 4) {
    float we = weight[e];
    int s = src[e], d = dst[e];
    float a  = (we >= 0.0f) ? s_ap[j] : s_an[j];
    float wj = 1.0f + __expf(we * a - s_cm[j]) * s_inv[j];
    atomicAdd(neigh + (size_t)d * DOUT + j, h1[(size_t)s * DOUT + j] * wj);
  }
}

// ---------------- fused MLP (64->128 relu ->64) + BN-stat epilogue ---------
__global__ void __launch_bounds__(128) k_mlp(const float* __restrict__ W1,
                                             const float* __restrict__ B1,
                                             const float* __restrict__ W2,
                                             const float* __restrict__ B2,
                                             float* ws) {
  const float* neigh = ws;
  float* h1 = ws + OFF_H1;  // read h2 = h1+neigh, then overwrite same rows w/ h3
  float* small = ws + OFF_SMALL;
  __shared__ float t[16][132];  // 132 stride: conflict-free column reads
  int lane = threadIdx.x & 31, wave = threadIdx.x >> 5;
  int m = lane & 15;
  int khi = (lane >= 16) ? 2 : 0;
  int rowbase = blockIdx.x * 16;
  // phase 1: hidden = relu(h2 @ W1^T + b1), two 16-col tiles per wave
  {
    v8f c0 = {}, c1 = {};
    int n0 = wave * 32, n1 = n0 + 16;
    size_t arow = (size_t)(rowbase + m) * DOUT;
    const float* b0row = W1 + (size_t)(n0 + m) * DOUT;
    const float* b1row = W1 + (size_t)(n1 + m) * DOUT;
#pragma unroll
    for (int k0 = 0; k0 < DOUT; k0 += 4) {
      int ka = k0 + khi;
      v2f a = *(const v2f*)(&h1[arow + ka]);
      v2f g = *(const v2f*)(&neigh[arow + ka]);
      a = a + g;  // h2 on the fly
      v2f bb0 = *(const v2f*)(b0row + ka);
      v2f bb1 = *(const v2f*)(b1row + ka);
      c0 = __builtin_amdgcn_wmma_f32_16x16x4_f32(false, a, false, bb0, (short)0, c0, false, false);
      c1 = __builtin_amdgcn_wmma_f32_16x16x4_f32(false, a, false, bb1, (short)0, c1, false, false);
    }
    int rr = (lane >= 16) ? 8 : 0;
    float bias0 = B1[n0 + m], bias1 = B1[n1 + m];
#pragma unroll
    for (int r = 0; r < 8; ++r) {
      t[rr + r][n0 + m] = fmaxf(c0[r] + bias0, 0.0f);
      t[rr + r][n1 + m] = fmaxf(c1[r] + bias1, 0.0f);
    }
  }
  __syncthreads();
  // phase 2: h3 = hidden @ W2^T + b2 (K=128 from LDS), + BN sum/sumsq
  {
    v8f c = {};
    int n0 = wave * 16;
    const float* brow = W2 + (size_t)(n0 + m) * HID;
#pragma unroll
    for (int k0 = 0; k0 < HID; k0 += 4) {
      int ka = k0 + khi;
      v2f a; a.x = t[m][ka]; a.y = t[m][ka + 1];
      v2f bb = *(const v2f*)(brow + ka);
      c = __builtin_amdgcn_wmma_f32_16x16x4_f32(false, a, false, bb, (short)0, c, false, false);
    }
    int rr = (lane >= 16) ? 8 : 0;
    int col = n0 + m;
    float bias = B2[col];
    float s = 0.0f, ss = 0.0f;
#pragma unroll
    for (int r = 0; r < 8; ++r) {
      float v = c[r] + bias;
      h1[(size_t)(rowbase + rr + r) * DOUT + col] = v;  // h3 in place
      s += v; ss += v * v;
    }
    s  += __shfl_xor(s, 16, 32);   // combine row halves (same column)
    ss += __shfl_xor(ss, 16, 32);
    if (lane < 16) {
      atomicAdd(small + 256 + col, s);
      atomicAdd(small + 320 + col, ss);
    }
  }
}

// ---------------- batchnorm + relu + residual ------------------------------
__global__ void __launch_bounds__(256) k_final(const float* __restrict__ h_in,
                                               const float* __restrict__ gamma,
                                               const float* __restrict__ beta,
                                               const float* ws, float* out) {
  const float* h3 = ws + OFF_H1;
  const float* small = ws + OFF_SMALL;
  const float invN = 1.0f / (float)N_NODES;
  size_t stride = (size_t)gridDim.x * blockDim.x;
  for (size_t i = (size_t)blockIdx.x * blockDim.x + threadIdx.x;
       i < (size_t)N_NODES * DOUT; i += stride) {
    int j = (int)(i & 63);
    float mean = small[256 + j] * invN;
    float var  = small[320 + j] * invN - mean * mean;  // biased var (ddof=0)
    float x = (h3[i] - mean) * rsqrtf(var + BN_EPS) * gamma[j] + beta[j];
    out[i] = h_in[i] + fmaxf(x, 0.0f);
  }
}

extern "C" void kernel_launch(void* const* d_in, const int* in_sizes, int n_in,
                              void* d_out, int out_size, void* d_ws, size_t ws_size,
                              hipStream_t stream) {
  const float* h      = (const float*)d_in[0];
  const float* weight = (const float*)d_in[1];
  const int*   src    = (const int*)d_in[2];
  const int*   dst    = (const int*)d_in[3];
  const float* lin_W  = (const float*)d_in[4];
  const float* lin_b  = (const float*)d_in[5];
  const float* w1_W   = (const float*)d_in[6];
  const float* w2_W   = (const float*)d_in[7];
  // d_in[8] = w2_b: constant per softmax column over edges -> cancels exactly
  const float* mlp_W1 = (const float*)d_in[9];
  const float* mlp_b1 = (const float*)d_in[10];
  const float* mlp_W2 = (const float*)d_in[11];
  const float* mlp_b2 = (const float*)d_in[12];
  const float* bn_g   = (const float*)d_in[13];
  const float* bn_b   = (const float*)d_in[14];
  float* ws  = (float*)d_ws;
  float* out = (float*)d_out;

  k_init   <<<512, 256, 0, stream>>>(ws);
  k_wred   <<<512, 256, 0, stream>>>(weight, ws);
  k_prep   <<<1, 64, 0, stream>>>(w1_W, w2_W, ws);
  k_lin    <<<(N_NODES / 16) * (DOUT / 16) / 4, 128, 0, stream>>>(h, lin_W, lin_b, ws);
  k_colsum <<<1024, 256, 0, stream>>>(weight, ws);
  k_scatter<<<4096, 256, 0, stream>>>(weight, src, dst, ws);
  k_mlp    <<<N_NODES / 16, 128, 0, stream>>>(mlp_W1, mlp_b1, mlp_W2, mlp_b2, ws);
  k_final  <<<2048, 256, 0, stream>>>(h, bn_g, bn_b, ws, out);
}